// LocalTypicalityMILHead_86509231276707
// MI455X (gfx1250) — compile-verified
//
#include <hip/hip_runtime.h>
#include <hip/hip_bf16.h>
#include <math.h>

#define N_INST 6144
#define IN_DIM 1024
#define HID_DIM 512
#define K_SEL 10
#define EPS_TH 19.0f

typedef __attribute__((ext_vector_type(2))) float v2f;
typedef __attribute__((ext_vector_type(8))) float v8f;

__device__ __forceinline__ v8f wmma_f32(v2f a, v2f b, v8f c) {
  // fp32 WMMA, K=4: (neg_a, A, neg_b, B, c_mod, C, reuse_a, reuse_b)
  return __builtin_amdgcn_wmma_f32_16x16x4_f32(false, a, false, b, (short)0, c,
                                               false, false);
}

// Async global->LDS copy (CDNA5): vdst = per-lane LDS byte address,
// voff = per-lane 32-bit byte offset, base = 64-bit SGPR base. ASYNCcnt.
__device__ __forceinline__ void async_copy_b128(unsigned ldsAddr, unsigned off,
                                                unsigned long long base) {
  asm volatile("global_load_async_to_lds_b128 %0, %1, %2"
               :: "v"(ldsAddr), "v"(off), "s"(base) : "memory");
}
__device__ __forceinline__ void async_wait0() {
  asm volatile("s_wait_asynccnt 0x0" ::: "memory");
}
__device__ __forceinline__ unsigned lds_addr(const float* p) {
  return (unsigned)(size_t)p;  // flat shared ptr: addr[31:0] == LDS offset
}

// ---------------------------------------------------------------------------
// Kernel 1: feats = relu(x @ W1 + b1)   [6144x1024 @ 1024x512]
// Block tile 64x128, 8 waves, each wave owns a 32x32 patch (2x2 WMMA block).
// A and B tiles staged via global_load_async_to_lds_b128.
// ---------------------------------------------------------------------------
__global__ __launch_bounds__(256) void gemm_feats_kernel(
    const float* __restrict__ x, const float* __restrict__ W1,
    const float* __restrict__ b1, float* __restrict__ feats) {
  __shared__ float As[64 * 36];   // 64 rows x 32 K (+4 pad)
  __shared__ float Bs[32 * 132];  // 32 K x 128 cols (+4 pad)
  const int tid = threadIdx.x;
  const int wave = tid >> 5, lane = tid & 31;
  const int half = lane >> 4, lr = lane & 15;
  const int m0 = blockIdx.y * 64;
  const int n0 = blockIdx.x * 128;
  const int wr = (wave >> 2) * 32;  // 0 or 32
  const int wc = (wave & 3) * 32;   // 0,32,64,96
  v8f c00 = {}, c01 = {}, c10 = {}, c11 = {};

  const int ar = tid >> 2, ac = (tid & 3) * 8;   // A: 2x b128 per thread
  const int bk = tid >> 3, bn = (tid & 7) * 16;  // B: 4x b128 per thread

  const unsigned ldsA = lds_addr(&As[ar * 36 + ac]);
  const unsigned ldsB = lds_addr(&Bs[bk * 132 + bn]);
  const unsigned long long baseX = (unsigned long long)(size_t)x;
  const unsigned long long baseW = (unsigned long long)(size_t)W1;

  for (int k0 = 0; k0 < IN_DIM; k0 += 32) {
    const unsigned offA = (unsigned)(((m0 + ar) * IN_DIM + k0 + ac) * 4);
    const unsigned offB = (unsigned)(((k0 + bk) * HID_DIM + n0 + bn) * 4);
    async_copy_b128(ldsA, offA, baseX);
    async_copy_b128(ldsA + 16u, offA + 16u, baseX);
    async_copy_b128(ldsB, offB, baseW);
    async_copy_b128(ldsB + 16u, offB + 16u, baseW);
    async_copy_b128(ldsB + 32u, offB + 32u, baseW);
    async_copy_b128(ldsB + 48u, offB + 48u, baseW);
    async_wait0();
    __syncthreads();

    for (int kk = 0; kk < 32; kk += 4) {
      const int ka = kk + 2 * half;
      v2f a0, a1, b0, b1v;
      a0[0] = As[(wr + lr) * 36 + ka + 0];
      a0[1] = As[(wr + lr) * 36 + ka + 1];
      a1[0] = As[(wr + 16 + lr) * 36 + ka + 0];
      a1[1] = As[(wr + 16 + lr) * 36 + ka + 1];
      b0[0] = Bs[(ka + 0) * 132 + wc + lr];
      b0[1] = Bs[(ka + 1) * 132 + wc + lr];
      b1v[0] = Bs[(ka + 0) * 132 + wc + 16 + lr];
      b1v[1] = Bs[(ka + 1) * 132 + wc + 16 + lr];
      c00 = wmma_f32(a0, b0, c00);
      c01 = wmma_f32(a0, b1v, c01);
      c10 = wmma_f32(a1, b0, c10);
      c11 = wmma_f32(a1, b1v, c11);
    }
    __syncthreads();
  }

  const int colA = n0 + wc + lr;
  const int colB = colA + 16;
  const float biasA = b1[colA];
  const float biasB = b1[colB];
#pragma unroll
  for (int v = 0; v < 8; ++v) {
    const int r0 = m0 + wr + v + 8 * half;
    const int r1 = r0 + 16;
    float f;
    f = c00[v] + biasA; f = f > 0.0f ? f : 0.0f;
    feats[(size_t)r0 * HID_DIM + colA] = f;
    f = c01[v] + biasB; f = f > 0.0f ? f : 0.0f;
    feats[(size_t)r0 * HID_DIM + colB] = f;
    f = c10[v] + biasA; f = f > 0.0f ? f : 0.0f;
    feats[(size_t)r1 * HID_DIM + colA] = f;
    f = c11[v] + biasB; f = f > 0.0f ? f : 0.0f;
    feats[(size_t)r1 * HID_DIM + colB] = f;
  }
}

// ---------------------------------------------------------------------------
// Kernel 2: sq[i] = sum(feats[i]^2); logits = feats @ Wc + bc  (one wave/row)
// ---------------------------------------------------------------------------
__global__ __launch_bounds__(256) void sq_logits_kernel(
    const float* __restrict__ feats, const float* __restrict__ Wc,
    const float* __restrict__ bc, float* __restrict__ sq,
    float* __restrict__ outLogits) {
  const int wave = threadIdx.x >> 5, lane = threadIdx.x & 31;
  const int row = blockIdx.x * 8 + wave;
  const float* f = feats + (size_t)row * HID_DIM;
  float s = 0.0f, l0 = 0.0f, l1 = 0.0f;
  for (int c = lane; c < HID_DIM; c += 32) {
    const float v = f[c];
    s += v * v;
    l0 += v * Wc[c * 2 + 0];
    l1 += v * Wc[c * 2 + 1];
  }
#pragma unroll
  for (int off = 16; off > 0; off >>= 1) {
    s += __shfl_xor(s, off, 32);
    l0 += __shfl_xor(l0, off, 32);
    l1 += __shfl_xor(l1, off, 32);
  }
  if (lane == 0) {
    sq[row] = s;
    outLogits[2 * row + 0] = l0 + bc[0];
    outLogits[2 * row + 1] = l1 + bc[1];
  }
}

// ---------------------------------------------------------------------------
// Kernel 3: G = feats @ feats^T ; D = sqrt(max(sq_i+sq_j-2G,1e-12))
// Block tile 64x128, 2x2 WMMA blocking. A tile async->LDS; B tile needs a
// register transpose so it keeps the VGPR path. fp64 atomics for sum stats.
// ---------------------------------------------------------------------------
__global__ __launch_bounds__(256) void gram_kernel(
    const float* __restrict__ feats, const float* __restrict__ sq,
    float* __restrict__ D, double* __restrict__ sums) {
  __shared__ float As[64 * 36];
  __shared__ float Bs[32 * 132];
  __shared__ float rs[256];
  __shared__ float rs2[256];
  const int tid = threadIdx.x;
  const int wave = tid >> 5, lane = tid & 31;
  const int half = lane >> 4, lr = lane & 15;
  const int m0 = blockIdx.y * 64;
  const int n0 = blockIdx.x * 128;
  const int wr = (wave >> 2) * 32;
  const int wc = (wave & 3) * 32;
  v8f c00 = {}, c01 = {}, c10 = {}, c11 = {};

  const int ar = tid >> 2, ac = (tid & 3) * 8;
  const int bn = tid >> 1;         // transposed-B source row (n): 0..127
  const int bk = (tid & 1) * 16;   // transposed-B k base: 0 or 16

  const unsigned ldsA = lds_addr(&As[ar * 36 + ac]);
  const unsigned long long baseF = (unsigned long long)(size_t)feats;

  for (int k0 = 0; k0 < HID_DIM; k0 += 32) {
    const unsigned offA = (unsigned)(((m0 + ar) * HID_DIM + k0 + ac) * 4);
    async_copy_b128(ldsA, offA, baseF);
    async_copy_b128(ldsA + 16u, offA + 16u, baseF);

    // B transpose into LDS: Bs[kk][n] = feats[n0+n][k0+kk]
    const float4* pb =
        (const float4*)(feats + (size_t)(n0 + bn) * HID_DIM + k0 + bk);
    float4 q0 = pb[0], q1 = pb[1], q2 = pb[2], q3 = pb[3];
    if (k0 + 32 < HID_DIM) {
      __builtin_prefetch(feats + (size_t)(n0 + bn) * HID_DIM + k0 + 32 + bk, 0, 0);
    }
    Bs[(bk + 0) * 132 + bn] = q0.x;  Bs[(bk + 1) * 132 + bn] = q0.y;
    Bs[(bk + 2) * 132 + bn] = q0.z;  Bs[(bk + 3) * 132 + bn] = q0.w;
    Bs[(bk + 4) * 132 + bn] = q1.x;  Bs[(bk + 5) * 132 + bn] = q1.y;
    Bs[(bk + 6) * 132 + bn] = q1.z;  Bs[(bk + 7) * 132 + bn] = q1.w;
    Bs[(bk + 8) * 132 + bn] = q2.x;  Bs[(bk + 9) * 132 + bn] = q2.y;
    Bs[(bk + 10) * 132 + bn] = q2.z; Bs[(bk + 11) * 132 + bn] = q2.w;
    Bs[(bk + 12) * 132 + bn] = q3.x; Bs[(bk + 13) * 132 + bn] = q3.y;
    Bs[(bk + 14) * 132 + bn] = q3.z; Bs[(bk + 15) * 132 + bn] = q3.w;
    async_wait0();
    __syncthreads();

    for (int kk = 0; kk < 32; kk += 4) {
      const int ka = kk + 2 * half;
      v2f a0, a1, b0, b1v;
      a0[0] = As[(wr + lr) * 36 + ka + 0];
      a0[1] = As[(wr + lr) * 36 + ka + 1];
      a1[0] = As[(wr + 16 + lr) * 36 + ka + 0];
      a1[1] = As[(wr + 16 + lr) * 36 + ka + 1];
      b0[0] = Bs[(ka + 0) * 132 + wc + lr];
      b0[1] = Bs[(ka + 1) * 132 + wc + lr];
      b1v[0] = Bs[(ka + 0) * 132 + wc + 16 + lr];
      b1v[1] = Bs[(ka + 1) * 132 + wc + 16 + lr];
      c00 = wmma_f32(a0, b0, c00);
      c01 = wmma_f32(a0, b1v, c01);
      c10 = wmma_f32(a1, b0, c10);
      c11 = wmma_f32(a1, b1v, c11);
    }
    __syncthreads();
  }

  const int colA = n0 + wc + lr;
  const int colB = colA + 16;
  const float sqA = sq[colA];
  const float sqB = sq[colB];
  float ls = 0.0f, ls2 = 0.0f;
#pragma unroll
  for (int v = 0; v < 8; ++v) {
    const int r0 = m0 + wr + v + 8 * half;
    const int r1 = r0 + 16;
    const float s0 = sq[r0];
    const float s1 = sq[r1];
    float d2, d;
    d2 = s0 + sqA - 2.0f * c00[v]; d2 = d2 > 1e-12f ? d2 : 1e-12f;
    d = sqrtf(d2); D[(size_t)r0 * N_INST + colA] = d; ls += d; ls2 += d * d;
    d2 = s0 + sqB - 2.0f * c01[v]; d2 = d2 > 1e-12f ? d2 : 1e-12f;
    d = sqrtf(d2); D[(size_t)r0 * N_INST + colB] = d; ls += d; ls2 += d * d;
    d2 = s1 + sqA - 2.0f * c10[v]; d2 = d2 > 1e-12f ? d2 : 1e-12f;
    d = sqrtf(d2); D[(size_t)r1 * N_INST + colA] = d; ls += d; ls2 += d * d;
    d2 = s1 + sqB - 2.0f * c11[v]; d2 = d2 > 1e-12f ? d2 : 1e-12f;
    d = sqrtf(d2); D[(size_t)r1 * N_INST + colB] = d; ls += d; ls2 += d * d;
  }
  rs[tid] = ls;
  rs2[tid] = ls2;
  __syncthreads();
  for (int s = 128; s > 0; s >>= 1) {
    if (tid < s) { rs[tid] += rs[tid + s]; rs2[tid] += rs2[tid + s]; }
    __syncthreads();
  }
  if (tid == 0) {
    atomicAdd(&sums[0], (double)rs[0]);
    atomicAdd(&sums[1], (double)rs2[0]);
  }
}

// ---------------------------------------------------------------------------
// Kernel 4: bandwidth h from global sums (sample std, ddof=1)
// ---------------------------------------------------------------------------
__global__ void finalize_h_kernel(const double* __restrict__ sums,
                                  float* __restrict__ hOut) {
  const double M = (double)N_INST * (double)N_INST;
  const double mean = sums[0] / M;
  double var = (sums[1] - M * mean * mean) / (M - 1.0);
  if (var < 0.0) var = 0.0;
  double s = sqrt(var);
  if (s < 1e-6) s = mean + 1e-6;
  double h = 1.06 * s * pow((double)N_INST, -0.2);
  if (h < 1e-3) h = 1e-3;
  hOut[0] = (float)h;
}

// ---------------------------------------------------------------------------
// Kernel 5: typ[i] = cnt>0 ? sum_{d<EPS} exp(-d^2/(2h^2)) / cnt : 0
// ---------------------------------------------------------------------------
__global__ __launch_bounds__(256) void typ_kernel(
    const float* __restrict__ D, const float* __restrict__ hPtr,
    float* __restrict__ typ, float* __restrict__ outTyp) {
  __shared__ float sks[256];
  __shared__ int scn[256];
  const int i = blockIdx.x;
  const int tid = threadIdx.x;
  const float h = hPtr[0];
  const float inv2h2 = 1.0f / (2.0f * h * h);
  const float* Drow = D + (size_t)i * N_INST;
  float ks = 0.0f;
  int cnt = 0;
  for (int col = tid; col < N_INST; col += 256) {
    const float d = Drow[col];
    if (d < EPS_TH) { cnt++; ks += expf(-d * d * inv2h2); }
  }
  sks[tid] = ks; scn[tid] = cnt;
  __syncthreads();
  for (int s = 128; s > 0; s >>= 1) {
    if (tid < s) { sks[tid] += sks[tid + s]; scn[tid] += scn[tid + s]; }
    __syncthreads();
  }
  if (tid == 0) {
    const float t = scn[0] > 0 ? sks[0] / (float)scn[0] : 0.0f;
    typ[i] = t;
    outTyp[i] = t;
  }
}

// ---------------------------------------------------------------------------
// Kernel 6: stable top-K selections (pos, far, neg). One block.
// ---------------------------------------------------------------------------
__global__ __launch_bounds__(256) void select_kernel(
    const float* __restrict__ D, const float* __restrict__ typ,
    const int* __restrict__ bagLabel, int* __restrict__ selIdx) {
  __shared__ float sv[256];
  __shared__ int si[256];
  __shared__ int posIdx[K_SEL];
  __shared__ int farIdx[K_SEL * K_SEL];
  __shared__ int negIdx[K_SEL];
  const int tid = threadIdx.x;
  const float sgn = (bagLabel[0] == 1) ? 1.0f : -1.0f;
  const float NEG_INF = -__builtin_inff();

  // --- pos: top-K of sgn*typ (stable: ties -> lowest index) ---
  for (int k = 0; k < K_SEL; ++k) {
    float best = NEG_INF; int bi = 0x7fffffff;
    for (int col = tid; col < N_INST; col += 256) {
      bool excl = false;
      for (int j = 0; j < k; ++j) excl |= (posIdx[j] == col);
      if (excl) continue;
      const float v = sgn * typ[col];
      if (v > best || (v == best && col < bi)) { best = v; bi = col; }
    }
    sv[tid] = best; si[tid] = bi;
    __syncthreads();
    for (int s = 128; s > 0; s >>= 1) {
      if (tid < s) {
        if (sv[tid + s] > sv[tid] ||
            (sv[tid + s] == sv[tid] && si[tid + s] < si[tid])) {
          sv[tid] = sv[tid + s]; si[tid] = si[tid + s];
        }
      }
      __syncthreads();
    }
    if (tid == 0) posIdx[k] = si[0];
    __syncthreads();
  }

  // --- far: per pos row, top-K of (posMember ? -1 : D[p,col]) ---
  for (int r = 0; r < K_SEL; ++r) {
    const float* Drow = D + (size_t)posIdx[r] * N_INST;
    for (int k = 0; k < K_SEL; ++k) {
      float best = NEG_INF; int bi = 0x7fffffff;
      for (int col = tid; col < N_INST; col += 256) {
        bool excl = false;
        for (int j = 0; j < k; ++j) excl |= (farIdx[r * K_SEL + j] == col);
        if (excl) continue;
        bool pm = false;
        for (int j = 0; j < K_SEL; ++j) pm |= (posIdx[j] == col);
        const float v = pm ? -1.0f : Drow[col];
        if (v > best || (v == best && col < bi)) { best = v; bi = col; }
      }
      sv[tid] = best; si[tid] = bi;
      __syncthreads();
      for (int s = 128; s > 0; s >>= 1) {
        if (tid < s) {
          if (sv[tid + s] > sv[tid] ||
              (sv[tid + s] == sv[tid] && si[tid + s] < si[tid])) {
            sv[tid] = sv[tid + s]; si[tid] = si[tid + s];
          }
        }
        __syncthreads();
      }
      if (tid == 0) farIdx[r * K_SEL + k] = si[0];
      __syncthreads();
    }
  }

  // --- neg: top-K of (-sgn*typ) over cand = union(far) \ pos ---
  for (int k = 0; k < K_SEL; ++k) {
    float best = NEG_INF; int bi = 0x7fffffff;
    for (int col = tid; col < N_INST; col += 256) {
      bool excl = false;
      for (int j = 0; j < k; ++j) excl |= (negIdx[j] == col);
      if (excl) continue;
      bool cand = false;
      for (int j = 0; j < K_SEL * K_SEL; ++j) cand |= (farIdx[j] == col);
      for (int j = 0; j < K_SEL; ++j) if (posIdx[j] == col) cand = false;
      const float v = cand ? (-sgn * typ[col]) : NEG_INF;
      if (v > best || (v == best && col < bi)) { best = v; bi = col; }
    }
    sv[tid] = best; si[tid] = bi;
    __syncthreads();
    for (int s = 128; s > 0; s >>= 1) {
      if (tid < s) {
        if (sv[tid + s] > sv[tid] ||
            (sv[tid + s] == sv[tid] && si[tid + s] < si[tid])) {
          sv[tid] = sv[tid + s]; si[tid] = si[tid + s];
        }
      }
      __syncthreads();
    }
    if (tid == 0) negIdx[k] = si[0];
    __syncthreads();
  }

  if (tid < K_SEL) {
    selIdx[tid] = posIdx[tid];
    selIdx[K_SEL + tid] = negIdx[tid];
  }
}

// ---------------------------------------------------------------------------
// Kernel 7: soft labels per row
// ---------------------------------------------------------------------------
__global__ __launch_bounds__(256) void soft_kernel(
    const float* __restrict__ D, const float* __restrict__ typ,
    const int* __restrict__ selIdx, float* __restrict__ outSoft) {
  const int i = blockIdx.x * 256 + threadIdx.x;
  if (i >= N_INST) return;
  const float* Drow = D + (size_t)i * N_INST;
  float pw = 0.0f, nw = 0.0f, ps = 0.0f, ns = 0.0f;
  int cm = 0;
#pragma unroll
  for (int j = 0; j < K_SEL; ++j) {
    const int pj = selIdx[j];
    const int nj = selIdx[K_SEL + j];
    const float dp = Drow[pj];
    const float dn = Drow[nj];
    if (dp < EPS_TH) { pw += typ[pj]; cm++; }
    if (dn < EPS_TH) { nw += typ[nj]; cm++; }
    ps += expf(-dp);
    ns += expf(-dn);
  }
  ps *= (1.0f / K_SEL);
  ns *= (1.0f / K_SEL);
  const float tot = pw + nw;
  const bool use_w = (cm > 0) && (tot > 0.0f);
  const float p = pw / (tot > 0.0f ? tot : 1.0f);
  const float t2 = ps + ns + 1e-6f;
  outSoft[2 * i + 0] = use_w ? (1.0f - p) : (ns / t2);
  outSoft[2 * i + 1] = use_w ? p : (ps / t2);
}

__global__ void init_kernel(double* sums) {
  if (threadIdx.x == 0) { sums[0] = 0.0; sums[1] = 0.0; }
}

// ---------------------------------------------------------------------------
extern "C" void kernel_launch(void* const* d_in, const int* in_sizes, int n_in,
                              void* d_out, int out_size, void* d_ws,
                              size_t ws_size, hipStream_t stream) {
  const float* x = (const float*)d_in[0];
  const float* W1 = (const float*)d_in[1];
  const float* b1 = (const float*)d_in[2];
  const float* Wc = (const float*)d_in[3];
  const float* bc = (const float*)d_in[4];
  const int* bag = (const int*)d_in[5];
  float* out = (float*)d_out;  // [logits 2N | soft 2N | typ N]
  char* ws = (char*)d_ws;

  const size_t offD = 0;
  const size_t offF = offD + (size_t)N_INST * N_INST * sizeof(float);
  const size_t offSq = offF + (size_t)N_INST * HID_DIM * sizeof(float);
  const size_t offTyp = offSq + (size_t)N_INST * sizeof(float);
  const size_t offS = (offTyp + (size_t)N_INST * sizeof(float) + 255) & ~(size_t)255;

  float* D = (float*)(ws + offD);
  float* feats = (float*)(ws + offF);
  float* sq = (float*)(ws + offSq);
  float* typ = (float*)(ws + offTyp);
  double* sums = (double*)(ws + offS);
  float* hPtr = (float*)(ws + offS + 16);
  int* selIdx = (int*)(ws + offS + 32);

  init_kernel<<<1, 32, 0, stream>>>(sums);
  gemm_feats_kernel<<<dim3(HID_DIM / 128, N_INST / 64), 256, 0, stream>>>(
      x, W1, b1, feats);
  sq_logits_kernel<<<N_INST / 8, 256, 0, stream>>>(feats, Wc, bc, sq, out);
  gram_kernel<<<dim3(N_INST / 128, N_INST / 64), 256, 0, stream>>>(feats, sq,
                                                                   D, sums);
  finalize_h_kernel<<<1, 1, 0, stream>>>(sums, hPtr);
  typ_kernel<<<N_INST, 256, 0, stream>>>(D, hPtr, typ, out + 4 * N_INST);
  select_kernel<<<1, 256, 0, stream>>>(D, typ, bag, selIdx);
  soft_kernel<<<(N_INST + 255) / 256, 256, 0, stream>>>(D, typ, selIdx,
                                                        out + 2 * N_INST);
}